// IF_29171417874855
// MI455X (gfx1250) — compile-verified
//
#include <hip/hip_runtime.h>

// Integrate-and-fire scan over time, CDNA5 (gfx1250) async-LDS streaming kernel.
//
// x: (T=32, B, N) f32 contiguous; y: same shape. Memory-bound (256 MB total
// traffic -> ~11 us floor at 23.3 TB/s). Each thread owns one float4 column.
// All 32 timesteps are prefetched with per-lane global_load_async_to_lds_b128
// (ASYNCcnt-tracked, zero VGPR landing zones, in-order completion), then
// consumed with s_wait_asynccnt.
//
// Addressing trick: the async-load INST_OFFSET immediate is added to BOTH the
// LDS address (VDST + off) and the global address (SADDR + VADDR + off), so we
// put the per-t LDS slot delta (t*2048) in the immediate and fold the residual
// global delta t*(gstep-2048) into the scalar base -> the two VGPR operands are
// loop-invariant and all per-t address math is SALU.

typedef float v4f __attribute__((ext_vector_type(4)));

#define TSTEPS 32
#define NT     128            // threads per block (4 waves of 32)
#define SLOT   (NT * 16)      // LDS bytes per timestep slot (2048)

__global__ __launch_bounds__(NT) void if_scan_async_kernel(
    const float* __restrict__ x, float* __restrict__ y, int bn4) {
  // [t][tid] layout: lane stride 16B -> conflict-free ds_load_b128.
  __shared__ v4f stage[TSTEPS][NT];   // 64 KB -> 5 blocks (20 waves) per WGP

  const int tid = threadIdx.x;
  const int i4  = blockIdx.x * NT + tid;
  if (i4 >= bn4) return;

  const unsigned           gbyte0 = (unsigned)i4 * 16u;       // per-lane, invariant
  const long long          gstep  = (long long)bn4 * 16;      // bytes per timestep
  const unsigned long long xbase  = (unsigned long long)(const void*)x;

  // Low 32 bits of a flat shared pointer == DS byte address (shared aperture).
  const unsigned lds0 = (unsigned)(size_t)&stage[0][tid];     // per-lane, invariant

  // ---- Issue the entire 32-deep async pipeline (in-order, ASYNCcnt <= 32) ----
#pragma unroll
  for (int t = 0; t < TSTEPS; ++t) {
    const unsigned long long sbase =
        xbase + (unsigned long long)((long long)t * (gstep - (long long)SLOT));
    asm volatile("global_load_async_to_lds_b128 %0, %1, %2 offset:%3"
                 :: "v"(lds0), "v"(gbyte0), "s"(sbase), "i"(t * SLOT)
                 : "memory");
  }

  v4f v = {0.0f, 0.0f, 0.0f, 0.0f};

  // ---- Serial IF scan; one asynccnt wait covers two timesteps ----
#pragma unroll
  for (int t = 0; t < TSTEPS; t += 2) {
    asm volatile("s_wait_asynccnt %0" :: "i"(TSTEPS - 2 - t) : "memory");

#pragma unroll
    for (int k = 0; k < 2; ++k) {
      v4f m = v + stage[t + k][tid];        // ds_load_b128 (imm offset) + integrate
      v4f spk;
      spk.x = (m.x >= 1.0f) ? 1.0f : 0.0f;  // Heaviside threshold (v_th = 1)
      spk.y = (m.y >= 1.0f) ? 1.0f : 0.0f;
      spk.z = (m.z >= 1.0f) ? 1.0f : 0.0f;
      spk.w = (m.w >= 1.0f) ? 1.0f : 0.0f;
      v = m - spk;                          // soft reset

      // Per-t scalar base -> saddr-form store with loop-invariant vector offset.
      v4f* ypt = (v4f*)((char*)y + (size_t)(t + k) * (size_t)gstep);
      __builtin_nontemporal_store(spk, ypt + i4);  // streamed, never re-read
    }
  }
}

extern "C" void kernel_launch(void* const* d_in, const int* in_sizes, int n_in,
                              void* d_out, int out_size, void* d_ws, size_t ws_size,
                              hipStream_t stream) {
  const float* x = (const float*)d_in[0];
  float*       y = (float*)d_out;

  const int total = in_sizes[0];          // T*B*N
  const int bn    = total / TSTEPS;       // elements per timestep (B*N)
  const int bn4   = bn / 4;               // float4 columns per timestep

  const int blocks = (bn4 + NT - 1) / NT;
  if_scan_async_kernel<<<dim3(blocks), dim3(NT), 0, stream>>>(x, y, bn4);
}